// S4DKernel_9122510536956
// MI455X (gfx1250) — compile-verified
//
#include <hip/hip_runtime.h>

typedef __attribute__((ext_vector_type(2))) float v2f;
typedef __attribute__((ext_vector_type(8))) float v8f;

#define DM 128       // d_model
#define NS 64        // state size (K dim of the GEMM)
#define TSTEPS 4     // time steps per block (amortize LDS fill)
#define PW (DM + 16) // padded pair-row width: 144 pairs = 288 words = 32 mod 64
                     // -> wave32 b64 LDS reads hit all 64 banks exactly once

__global__ __launch_bounds__(256) void s4d_wmma_kernel(
    const float* __restrict__ Amat,   // (64,64), diagonal matters
    const float* __restrict__ B,      // (64,128)
    const float* __restrict__ C,      // (128,64)
    float* __restrict__ out,          // (L,128,128)
    int L)
{
    // K-pair-interleaved operands: fragment = one aligned ds_load_b64.
    // Bp[p][n] = (B[2p][n],   B[2p+1][n])
    // Cp[p][m] = (C[m][2p],   C[m][2p+1])
    __shared__ v2f Bp[NS / 2][PW];
    __shared__ v2f Cp[NS / 2][PW];
    __shared__ v2f slp[TSTEPS][NS / 2];   // (a[2p]^i, a[2p+1]^i)

    float* Bpf  = (float*)Bp;
    float* Cpf  = (float*)Cp;
    float* slpf = (float*)slp;

    const int tid = threadIdx.x;
    const int i0  = blockIdx.x * TSTEPS;

    // Cooperative LDS fill (inputs stay resident in L2 across blocks).
    for (int t = tid; t < NS * DM; t += 256) {
        const int k = t >> 7, n = t & 127;          // B row-major (64,128)
        Bpf[((k >> 1) * PW + n) * 2 + (k & 1)] = B[t];
    }
    for (int t = tid; t < DM * NS; t += 256) {
        const int m = t >> 6, k = t & 63;           // C row-major (128,64)
        Cpf[((k >> 1) * PW + m) * 2 + (k & 1)] = C[t];
    }
    if (tid < NS) {
        const float a = Amat[tid * (NS + 1)];       // diagonal element a[k]
        float s = powf(a, (float)i0);               // a^i0 (powf(x,0)==1)
        #pragma unroll
        for (int t = 0; t < TSTEPS; ++t) {
            slpf[(t * (NS / 2) + (tid >> 1)) * 2 + (tid & 1)] = s;
            s *= a;
        }
    }
    __syncthreads();

    const int lane = tid & 31;
    const int wave = tid >> 5;        // 0..7 -> M strip of 16 rows
    const int half = lane >> 4;       // lane half selects K pair within K=4
    const int li   = lane & 15;
    const int Mb   = wave * 16;

    for (int t = 0; t < TSTEPS; ++t) {
        const int i = i0 + t;
        if (i >= L) break;            // uniform branch: EXEC stays all-ones

        v8f acc[8] = {};              // 8 N-tiles of 16x16 f32

        #pragma unroll
        for (int ks = 0; ks < 16; ++ks) {       // K = 64 in steps of 4
            const int p0 = ks * 2 + half;       // pair index (= kb/2)
            // A-fragment: (C * a^i), one b64 load + one v_pk_mul_f32
            const v2f af = Cp[p0][Mb + li] * slp[t][p0];
            #pragma unroll
            for (int nt = 0; nt < 8; ++nt) {
                const v2f bf = Bp[p0][nt * 16 + li];   // one b64 load
                acc[nt] = __builtin_amdgcn_wmma_f32_16x16x4_f32(
                    /*neg_a=*/false, af, /*neg_b=*/false, bf,
                    /*c_mod=*/(short)0, acc[nt],
                    /*reuse_a=*/false, /*reuse_b=*/false);
            }
        }

        // Store: VGPR r of tile nt holds (m = Mb + r + 8*half, n = nt*16+li).
        // Output is write-once and bigger than L2 -> non-temporal stores.
        float* obase = out + (size_t)i * DM * DM;
        #pragma unroll
        for (int nt = 0; nt < 8; ++nt) {
            #pragma unroll
            for (int r = 0; r < 8; ++r) {
                const int m = Mb + r + 8 * half;
                const int n = nt * 16 + li;
                __builtin_nontemporal_store(acc[nt][r],
                                            obase + (size_t)m * DM + n);
            }
        }
    }
}

extern "C" void kernel_launch(void* const* d_in, const int* in_sizes, int n_in,
                              void* d_out, int out_size, void* d_ws, size_t ws_size,
                              hipStream_t stream) {
    (void)in_sizes; (void)n_in; (void)d_ws; (void)ws_size;
    const float* A = (const float*)d_in[0];   // (64,64) f32
    const float* B = (const float*)d_in[1];   // (64,128) f32
    const float* C = (const float*)d_in[2];   // (128,64) f32
    float* out = (float*)d_out;               // (L,128,128) f32

    const int L = out_size / (DM * DM);       // L from output size (4096)
    const int blocks = (L + TSTEPS - 1) / TSTEPS;
    if (blocks > 0) {
        s4d_wmma_kernel<<<blocks, 256, 0, stream>>>(A, B, C, out, L);
    }
}